// RANSAC_52467320488586
// MI455X (gfx1250) — compile-verified
//
#include <hip/hip_runtime.h>
#include <hip/hip_bf16.h>
#include <math.h>

// ---------------------------------------------------------------------------
// RANSAC homography for MI455X (gfx1250, wave32).
// Dense hotspot (1024 models x 4096 points projective verification) runs
// through V_WMMA_F32_16X16X4_F32 (fp32 required: coords ~1024, inlier
// threshold err^2 <= 2.0, f16 quantization would exceed the threshold).
// This revision replaces IEEE-division / sqrt / exp expansions with the
// single-instruction hardware approximations (v_rcp_f32 / v_rsq_f32 /
// v_exp_f32): ~1 ulp accuracy, far beyond what the inlier test needs, and
// removes ~7 VALU ops per perspective divide in the WMMA loop.
// Deterministic: fixed kernel ladder, integer-valued float atomics only.
// ---------------------------------------------------------------------------

typedef __attribute__((ext_vector_type(2))) float v2f;
typedef __attribute__((ext_vector_type(8))) float v8f;

#define EPSF 1e-8f

__device__ __forceinline__ float frcp(float x)  { return __builtin_amdgcn_rcpf(x); }
__device__ __forceinline__ float frsq(float x)  { return __builtin_amdgcn_rsqf(x); }

constexpr int NPTS      = 4096;
constexpr int NBATCH    = 1024;
constexpr int RS_ITERS  = 10;
constexpr int RS_LO     = 5;
constexpr int RS_DLT    = 5;

// -------------------- workspace layout (float units) ------------------------
constexpr int MODELS  = 0;                    // 1024 * 9
constexpr int FLAGS   = MODELS + 9 * NBATCH;  // 1024 (valid && diag_ok)
constexpr int SCORES  = FLAGS + NBATCH;       // 1024
constexpr int ST      = SCORES + NBATCH;      // scalar state block
constexpr int S_BESTM  = ST + 0;   // 9
constexpr int S_BESTSC = ST + 9;
constexpr int S_DONE   = ST + 10;
constexpr int S_IMPROVE= ST + 11;
constexpr int S_CURM   = ST + 12;  // 9
constexpr int S_CURSC  = ST + 21;
constexpr int S_TRM    = ST + 22;  // 9
constexpr int S_TRSC   = ST + 31;
constexpr int S_ACTIVE = ST + 32;
constexpr int TPAR     = ST + 40;  // s1,mx1,my1,s2,mx2,my2
constexpr int P1N     = ST + 64;             // 2*4096 normalized kp1
constexpr int P2N     = P1N + 2 * NPTS;      // 2*4096 normalized kp2
constexpr int WBUF    = P2N + 2 * NPTS;      // 4096 DLT weights
constexpr int MASKB   = WBUF + NPTS;         // 4096 (w>0 mask)
constexpr int CURINL  = MASKB + NPTS;        // 4096 carry inliers
constexpr int TRINL   = CURINL + NPTS;       // 4096 trial inliers
constexpr int BESTINL = TRINL + NPTS;        // 4096 best inliers
constexpr int ATWAP   = BESTINL + NPTS;      // 32 * 45 partial AtWA

// -------------------- small device helpers ----------------------------------
__device__ __forceinline__ unsigned hash_u32(unsigned x) {
  x ^= x >> 16; x *= 0x7feb352dU;
  x ^= x >> 15; x *= 0x846ca68bU;
  x ^= x >> 16; return x;
}

__device__ __forceinline__ float det3h(float ax, float ay, float bx, float by,
                                       float cx, float cy) {
  // det [[ax,ay,1],[bx,by,1],[cx,cy,1]]
  return ax * (by - cy) - ay * (bx - cx) + (bx * cy - by * cx);
}

__device__ __forceinline__ int sgnf(float v) { return (v > 0.f) - (v < 0.f); }

__device__ __forceinline__ float transfer_err2(const float H[9], float x1,
                                               float y1, float x2, float y2) {
  float u = H[0] * x1 + H[1] * y1 + H[2];
  float v = H[3] * x1 + H[4] * y1 + H[5];
  float z = H[6] * x1 + H[7] * y1 + H[8];
  float zi = (fabsf(z) > EPSF) ? frcp(z) : 1.f;   // v_rcp_f32
  float ex = u * zi - x2, ey = v * zi - y2;
  return ex * ex + ey * ey;
}

// cyclic Jacobi eigensolver for symmetric 9x9; eigvecs in columns of v
__device__ void jacobi9(float a[9][9], float v[9][9]) {
  for (int i = 0; i < 9; i++)
    for (int j = 0; j < 9; j++) v[i][j] = (i == j) ? 1.f : 0.f;
  for (int sweep = 0; sweep < 8; sweep++) {
    for (int p = 0; p < 8; p++) {
      for (int q = p + 1; q < 9; q++) {
        float apq = a[p][q];
        if (fabsf(apq) < 1e-24f) continue;
        float theta = (a[q][q] - a[p][p]) * 0.5f * frcp(apq);
        float t = frcp(fabsf(theta) + sqrtf(theta * theta + 1.f));
        if (theta < 0.f) t = -t;
        float c = frsq(t * t + 1.f);             // v_rsq_f32
        float s = t * c;
        for (int i = 0; i < 9; i++) {            // A = A * G
          float aip = a[i][p], aiq = a[i][q];
          a[i][p] = c * aip - s * aiq;
          a[i][q] = s * aip + c * aiq;
        }
        for (int i = 0; i < 9; i++) {            // A = G^T * A
          float api = a[p][i], aqi = a[q][i];
          a[p][i] = c * api - s * aqi;
          a[q][i] = s * api + c * aqi;
        }
        for (int i = 0; i < 9; i++) {            // V = V * G
          float vip = v[i][p], viq = v[i][q];
          v[i][p] = c * vip - s * viq;
          v[i][q] = s * vip + c * viq;
        }
      }
    }
  }
}

// H = inv(T2) * Hn * T1, then H /= (H[8] + EPS)
__device__ void denorm_h(const float hn[9], float s1, float mx1, float my1,
                         float s2, float mx2, float my2, float H[9]) {
  float M[9];
  for (int r = 0; r < 3; r++) {
    M[r * 3 + 0] = s1 * hn[r * 3 + 0];
    M[r * 3 + 1] = s1 * hn[r * 3 + 1];
    M[r * 3 + 2] = -s1 * mx1 * hn[r * 3 + 0] - s1 * my1 * hn[r * 3 + 1] + hn[r * 3 + 2];
  }
  float is2 = frcp(s2);
  for (int c = 0; c < 3; c++) {
    H[0 * 3 + c] = is2 * M[0 * 3 + c] + mx2 * M[2 * 3 + c];
    H[1 * 3 + c] = is2 * M[1 * 3 + c] + my2 * M[2 * 3 + c];
    H[2 * 3 + c] = M[2 * 3 + c];
  }
  float di = frcp(H[8] + EPSF);
  for (int j = 0; j < 9; j++) H[j] *= di;
}

// -------------------- init / normalization ----------------------------------
__global__ void init_state(float* ws) {
  int t = blockIdx.x * blockDim.x + threadIdx.x;
  if (t < NPTS) ws[BESTINL + t] = 0.f;
  if (t < 64) ws[ST + t] = (t == 9) ? 4.0f : 0.f;  // best_score = SAMPLE
}

__global__ void norm_reduce(const float* __restrict__ kp1,
                            const float* __restrict__ kp2, float* ws) {
  __shared__ float r0[256], r1[256], r2[256], r3[256];
  __shared__ float m[4];
  int t = threadIdx.x;
  float sx1 = 0, sy1 = 0, sx2 = 0, sy2 = 0;
  for (int n = t; n < NPTS; n += 256) {
    sx1 += kp1[2 * n]; sy1 += kp1[2 * n + 1];
    sx2 += kp2[2 * n]; sy2 += kp2[2 * n + 1];
  }
  r0[t] = sx1; r1[t] = sy1; r2[t] = sx2; r3[t] = sy2;
  __syncthreads();
  for (int s = 128; s > 0; s >>= 1) {
    if (t < s) { r0[t] += r0[t + s]; r1[t] += r1[t + s]; r2[t] += r2[t + s]; r3[t] += r3[t + s]; }
    __syncthreads();
  }
  if (t == 0) {
    const float inv = 1.f / (float)NPTS;
    m[0] = r0[0] * inv; m[1] = r1[0] * inv;
    m[2] = r2[0] * inv; m[3] = r3[0] * inv;
  }
  __syncthreads();
  float mx1 = m[0], my1 = m[1], mx2 = m[2], my2 = m[3];
  float d1 = 0, d2 = 0;
  for (int n = t; n < NPTS; n += 256) {
    float dx = kp1[2 * n] - mx1, dy = kp1[2 * n + 1] - my1;
    d1 += sqrtf(dx * dx + dy * dy);
    dx = kp2[2 * n] - mx2; dy = kp2[2 * n + 1] - my2;
    d2 += sqrtf(dx * dx + dy * dy);
  }
  r0[t] = d1; r1[t] = d2;
  __syncthreads();
  for (int s = 128; s > 0; s >>= 1) {
    if (t < s) { r0[t] += r0[t + s]; r1[t] += r1[t + s]; }
    __syncthreads();
  }
  if (t == 0) {
    const float inv = 1.f / (float)NPTS;
    float sc1 = r0[0] * inv, sc2 = r1[0] * inv;
    float rt2 = 1.41421356237309515f;
    ws[TPAR + 0] = rt2 * frcp(sc1 + EPSF); ws[TPAR + 1] = mx1; ws[TPAR + 2] = my1;
    ws[TPAR + 3] = rt2 * frcp(sc2 + EPSF); ws[TPAR + 4] = mx2; ws[TPAR + 5] = my2;
  }
}

__global__ void norm_apply(const float* __restrict__ kp1,
                           const float* __restrict__ kp2, float* ws) {
  int n = blockIdx.x * blockDim.x + threadIdx.x;
  if (n >= NPTS) return;
  float s1 = ws[TPAR + 0], mx1 = ws[TPAR + 1], my1 = ws[TPAR + 2];
  float s2 = ws[TPAR + 3], mx2 = ws[TPAR + 4], my2 = ws[TPAR + 5];
  ws[P1N + 2 * n]     = s1 * (kp1[2 * n] - mx1);
  ws[P1N + 2 * n + 1] = s1 * (kp1[2 * n + 1] - my1);
  ws[P2N + 2 * n]     = s2 * (kp2[2 * n] - mx2);
  ws[P2N + 2 * n + 1] = s2 * (kp2[2 * n + 1] - my2);
}

// -------------------- minimal 4-point DLT per model --------------------------
__global__ void gen_models(const float* __restrict__ kp1,
                           const float* __restrict__ kp2, float* ws, int iter) {
  int b = blockIdx.x * blockDim.x + threadIdx.x;
  if (b >= NBATCH) return;
  ws[SCORES + b] = 0.f;

  // counter-hash sampling of 4 distinct indices (deterministic per call)
  int idx[4];
  unsigned base = (unsigned)(iter * NBATCH + b);
  for (int k = 0; k < 4; k++) {
    int cand = (int)(hash_u32(base * 4u + (unsigned)k + 0x9E3779B9u) & (NPTS - 1));
    bool dup = true;
    while (dup) {
      dup = false;
      for (int j = 0; j < k; j++)
        if (idx[j] == cand) { cand = (cand + 1) & (NPTS - 1); dup = true; }
    }
    idx[k] = cand;
  }
  float x1[4], y1[4], x2[4], y2[4];
  for (int k = 0; k < 4; k++) {
    x1[k] = kp1[2 * idx[k]]; y1[k] = kp1[2 * idx[k] + 1];
    x2[k] = kp2[2 * idx[k]]; y2[k] = kp2[2 * idx[k] + 1];
  }
  // orientation consistency of all 3-point subsets
  const int tr[4][3] = {{0, 1, 2}, {0, 1, 3}, {0, 2, 3}, {1, 2, 3}};
  bool valid = true;
  for (int t = 0; t < 4; t++) {
    int a = tr[t][0], bb = tr[t][1], c = tr[t][2];
    int sA = sgnf(det3h(x1[a], y1[a], x1[bb], y1[bb], x1[c], y1[c]));
    int sB = sgnf(det3h(x2[a], y2[a], x2[bb], y2[bb], x2[c], y2[c]));
    if (sA != sB) valid = false;
  }
  // normalize the 4-point sets
  float mx1 = 0, my1 = 0, mx2 = 0, my2 = 0;
  for (int k = 0; k < 4; k++) { mx1 += x1[k]; my1 += y1[k]; mx2 += x2[k]; my2 += y2[k]; }
  mx1 *= 0.25f; my1 *= 0.25f; mx2 *= 0.25f; my2 *= 0.25f;
  float sc1 = 0, sc2 = 0;
  for (int k = 0; k < 4; k++) {
    float dx = x1[k] - mx1, dy = y1[k] - my1; sc1 += sqrtf(dx * dx + dy * dy);
    dx = x2[k] - mx2; dy = y2[k] - my2;       sc2 += sqrtf(dx * dx + dy * dy);
  }
  sc1 *= 0.25f; sc2 *= 0.25f;
  float rt2 = 1.41421356237309515f;
  float s1 = rt2 * frcp(sc1 + EPSF), s2 = rt2 * frcp(sc2 + EPSF);

  // AtA of the 8x9 DLT system (weights = 1)
  float A[9][9];
  for (int i = 0; i < 9; i++) for (int j = 0; j < 9; j++) A[i][j] = 0.f;
  for (int k = 0; k < 4; k++) {
    float X1 = s1 * (x1[k] - mx1), Y1 = s1 * (y1[k] - my1);
    float X2 = s2 * (x2[k] - mx2), Y2 = s2 * (y2[k] - my2);
    float ax[9] = {0, 0, 0, -X1, -Y1, -1.f, Y2 * X1, Y2 * Y1, Y2};
    float ay[9] = {X1, Y1, 1.f, 0, 0, 0, -X2 * X1, -X2 * Y1, -X2};
    for (int i = 0; i < 9; i++)
      for (int j = 0; j < 9; j++)
        A[i][j] += ax[i] * ax[j] + ay[i] * ay[j];
  }
  float V[9][9];
  jacobi9(A, V);
  int kmin = 0; float mn = A[0][0];
  for (int i = 1; i < 9; i++) if (A[i][i] < mn) { mn = A[i][i]; kmin = i; }
  float hn[9];
  for (int r = 0; r < 9; r++) hn[r] = V[r][kmin];
  float H[9];
  denorm_h(hn, s1, mx1, my1, s2, mx2, my2, H);
  bool diag_ok = fminf(fminf(fabsf(H[0]), fabsf(H[4])), fabsf(H[8])) > 1e-4f;
  for (int j = 0; j < 9; j++) ws[MODELS + b * 9 + j] = H[j];
  ws[FLAGS + b] = (valid && diag_ok) ? 1.f : 0.f;
}

// -------------------- WMMA batch verification --------------------------------
// Block = 256 threads = 8 waves; block owns a tile of 16 models, each wave
// scans 32 point-tiles of 16 points.  Per tile: 3x v_wmma_f32_16x16x4_f32
// producing u/v/z rows of all 16 models for 16 points at once; perspective
// divide via single v_rcp_f32 per row.
__global__ void verify_batch(const float* __restrict__ kp1,
                             const float* __restrict__ kp2, float* ws) {
  const int mt = blockIdx.x;                   // 0..63
  const int wave = threadIdx.x >> 5;           // 0..7
  const int lane = threadIdx.x & 31;
  const int col = lane & 15;                   // model column in tile
  const bool hi = lane >= 16;                  // K=2,3 half (ISA A/B layout)
  const int m = mt * 16 + col;
  const float* H = ws + MODELS + m * 9;

  // B tiles: column m holds one model row (K index = homog coordinate)
  v2f bu, bv, bz;
  if (!hi) { bu = v2f{H[0], H[1]}; bv = v2f{H[3], H[4]}; bz = v2f{H[6], H[7]}; }
  else     { bu = v2f{H[2], 0.f};  bv = v2f{H[5], 0.f};  bz = v2f{H[8], 0.f}; }

  float cnt = 0.f;
  for (int pt = wave; pt < NPTS / 16; pt += 8) {
    const int p = pt * 16 + col;               // A-tile row M = col
    v2f a;
    if (!hi) a = v2f{kp1[2 * p], kp1[2 * p + 1]};   // K0=x, K1=y
    else     a = v2f{1.f, 0.f};                      // K2=1, K3=0 (pad)
    v8f du = {}, dv = {}, dz = {};
    du = __builtin_amdgcn_wmma_f32_16x16x4_f32(false, a, false, bu, (short)0, du, false, false);
    dv = __builtin_amdgcn_wmma_f32_16x16x4_f32(false, a, false, bv, (short)0, dv, false, false);
    dz = __builtin_amdgcn_wmma_f32_16x16x4_f32(false, a, false, bz, (short)0, dz, false, false);
#pragma unroll
    for (int r = 0; r < 8; r++) {
      int pr = pt * 16 + r + (hi ? 8 : 0);     // D row M = r (+8 for hi half)
      float x2 = kp2[2 * pr], y2 = kp2[2 * pr + 1];
      float z = dz[r];
      float zi = (fabsf(z) > EPSF) ? frcp(z) : 1.f;  // v_rcp_f32, not IEEE div
      float ex = du[r] * zi - x2, ey = dv[r] * zi - y2;
      float e = ex * ex + ey * ey;
      cnt += (e <= 2.0f) ? 1.f : 0.f;
    }
  }
  cnt += __shfl_xor(cnt, 16, 32);              // combine row halves per column
  if (!hi) atomicAdd(&ws[SCORES + m], cnt);    // integer-valued: deterministic
}

// -------------------- argmax + carry bookkeeping -----------------------------
__global__ void select_best(float* ws) {
  __shared__ float sv[1024];
  __shared__ int si[1024];
  int t = threadIdx.x;
  float sc = ws[SCORES + t];
  sc = (ws[FLAGS + t] > 0.5f) ? sc : -1.f;
  sv[t] = sc; si[t] = t;
  __syncthreads();
  for (int s = 512; s > 0; s >>= 1) {
    if (t < s) {
      if (sv[t + s] > sv[t]) { sv[t] = sv[t + s]; si[t] = si[t + s]; }
    }
    __syncthreads();
  }
  if (t == 0) {
    int bi = si[0];
    float score = sv[0];
    bool improve = (ws[S_DONE] < 0.5f) && (score > ws[S_BESTSC]);
    ws[S_IMPROVE] = improve ? 1.f : 0.f;
    ws[S_ACTIVE] = improve ? 1.f : 0.f;
    for (int j = 0; j < 9; j++) ws[S_CURM + j] = ws[MODELS + bi * 9 + j];
    ws[S_CURSC] = score;
  }
}

__global__ void compute_mask(const float* __restrict__ kp1,
                             const float* __restrict__ kp2, float* ws) {
  int n = blockIdx.x * blockDim.x + threadIdx.x;
  if (n >= NPTS) return;
  float H[9];
  for (int j = 0; j < 9; j++) H[j] = ws[S_CURM + j];
  float e = transfer_err2(H, kp1[2 * n], kp1[2 * n + 1], kp2[2 * n], kp2[2 * n + 1]);
  float inl = (e <= 2.0f) ? 1.f : 0.f;
  ws[CURINL + n] = inl; ws[WBUF + n] = inl; ws[MASKB + n] = inl;
}

// -------------------- weighted DLT over all points ---------------------------
__global__ void atwa_partial(float* ws) {
  int tid = threadIdx.x;                       // 128 threads
  int n = blockIdx.x * 128 + tid;              // 32 blocks
  const float x1 = ws[P1N + 2 * n], y1 = ws[P1N + 2 * n + 1];
  const float x2 = ws[P2N + 2 * n], y2 = ws[P2N + 2 * n + 1];
  const float wn = ws[WBUF + n];
  float ax[9] = {0, 0, 0, -x1, -y1, -1.f, y2 * x1, y2 * y1, y2};
  float ay[9] = {x1, y1, 1.f, 0, 0, 0, -x2 * x1, -x2 * y1, -x2};
  float loc[45];
  {
    int e = 0;
#pragma unroll
    for (int i = 0; i < 9; i++)
#pragma unroll
      for (int j = i; j < 9; j++) { loc[e] = wn * (ax[i] * ax[j] + ay[i] * ay[j]); e++; }
  }
  __shared__ float red[128];
#pragma unroll
  for (int e = 0; e < 45; e++) {
    red[tid] = loc[e];
    __syncthreads();
    for (int s = 64; s > 0; s >>= 1) {
      if (tid < s) red[tid] += red[tid + s];
      __syncthreads();
    }
    if (tid == 0) ws[ATWAP + blockIdx.x * 45 + e] = red[0];
    __syncthreads();
  }
}

__global__ void atwa_eig(float* ws) {
  __shared__ float tri[45];
  int t = threadIdx.x;
  if (t < 45) {
    float s = 0.f;
    for (int b = 0; b < 32; b++) s += ws[ATWAP + b * 45 + t];  // fixed order
    tri[t] = s;
  }
  __syncthreads();
  if (t != 0) return;
  float A[9][9], V[9][9];
  int e = 0;
  for (int i = 0; i < 9; i++)
    for (int j = i; j < 9; j++) { A[i][j] = tri[e]; A[j][i] = tri[e]; e++; }
  jacobi9(A, V);
  int k = 0; float mn = A[0][0];
  for (int i = 1; i < 9; i++) if (A[i][i] < mn) { mn = A[i][i]; k = i; }
  float hn[9];
  for (int r = 0; r < 9; r++) hn[r] = V[r][k];
  float H[9];
  denorm_h(hn, ws[TPAR + 0], ws[TPAR + 1], ws[TPAR + 2],
               ws[TPAR + 3], ws[TPAR + 4], ws[TPAR + 5], H);
  for (int j = 0; j < 9; j++) ws[S_TRM + j] = H[j];
}

__global__ void reweight(const float* __restrict__ kp1,
                         const float* __restrict__ kp2, float* ws) {
  int n = blockIdx.x * blockDim.x + threadIdx.x;
  if (n >= NPTS) return;
  float H[9];
  for (int j = 0; j < 9; j++) H[j] = ws[S_TRM + j];
  float e2 = transfer_err2(H, kp1[2 * n], kp1[2 * n + 1], kp2[2 * n], kp2[2 * n + 1]);
  float err = sqrtf(e2 + EPSF);
  ws[WBUF + n] = ws[MASKB + n] * __expf(-err * (1.f / 18.f));  // v_exp_f32
}

__global__ void verify_single(const float* __restrict__ kp1,
                              const float* __restrict__ kp2, float* ws) {
  int n = blockIdx.x * blockDim.x + threadIdx.x;
  if (n >= NPTS) return;
  float H[9];
  for (int j = 0; j < 9; j++) H[j] = ws[S_TRM + j];
  float e = transfer_err2(H, kp1[2 * n], kp1[2 * n + 1], kp2[2 * n], kp2[2 * n + 1]);
  float inl = (e <= 2.0f) ? 1.f : 0.f;
  ws[TRINL + n] = inl;
  if (inl > 0.f) atomicAdd(&ws[S_TRSC], 1.f);         // exact integer counts
}

__global__ void set_scalar(float* p, float v) {
  if (threadIdx.x == 0 && blockIdx.x == 0) *p = v;
}

__global__ void polish_update(float* ws) {
  if (threadIdx.x != 0 || blockIdx.x != 0) return;
  bool better = (ws[S_ACTIVE] > 0.5f) && (ws[S_TRSC] > ws[S_CURSC]);
  if (better) {
    for (int j = 0; j < 9; j++) ws[S_CURM + j] = ws[S_TRM + j];
    ws[S_CURSC] = ws[S_TRSC];
  }
  ws[S_ACTIVE] = better ? 1.f : 0.f;
}

__global__ void post_lo(float* ws) {
  int n = blockIdx.x * blockDim.x + threadIdx.x;
  if (n >= NPTS) return;
  if (ws[S_ACTIVE] > 0.5f) ws[CURINL + n] = ws[TRINL + n];
  float c = ws[CURINL + n];
  ws[WBUF + n] = c; ws[MASKB + n] = c;
}

__global__ void iter_update(float* ws, int iter) {
  if (threadIdx.x != 0 || blockIdx.x != 0) return;
  bool improve = ws[S_IMPROVE] > 0.5f;
  if (improve) {
    for (int j = 0; j < 9; j++) ws[S_BESTM + j] = ws[S_CURM + j];
    ws[S_BESTSC] = ws[S_CURSC];
  }
  float bs = ws[S_BESTSC];
  float ratio = floorf(bs) / (float)NPTS;
  float r2 = ratio * ratio;
  float q = 1.f - r2 * r2;
  q = fminf(fmaxf(q, 1e-12f), 1.f - 1e-12f);
  float max_samp = (bs >= (float)NPTS) ? 1.f : (logf(0.01f) / logf(q));
  bool done = (ws[S_DONE] > 0.5f) ||
              (improve && ((float)((iter + 1) * NBATCH) >= floorf(max_samp)));
  ws[S_DONE] = done ? 1.f : 0.f;
}

__global__ void copy_best_inl(float* ws) {
  int n = blockIdx.x * blockDim.x + threadIdx.x;
  if (n >= NPTS) return;
  if (ws[S_IMPROVE] > 0.5f) ws[BESTINL + n] = ws[CURINL + n];
}

__global__ void write_out(const float* __restrict__ ws, float* __restrict__ out,
                          int out_size) {
  int t = blockIdx.x * blockDim.x + threadIdx.x;
  if (t >= out_size) return;
  if (t < 9) out[t] = ws[S_BESTM + t];
  else {
    int n = t - 9;
    out[t] = (n < NPTS) ? ws[BESTINL + n] : 0.f;
  }
}

// -------------------- host orchestration ------------------------------------
extern "C" void kernel_launch(void* const* d_in, const int* in_sizes, int n_in,
                              void* d_out, int out_size, void* d_ws, size_t ws_size,
                              hipStream_t stream) {
  (void)in_sizes; (void)n_in; (void)ws_size;
  const float* kp1 = (const float*)d_in[0];
  const float* kp2 = (const float*)d_in[1];
  float* out = (float*)d_out;
  float* ws = (float*)d_ws;

  init_state<<<17, 256, 0, stream>>>(ws);
  norm_reduce<<<1, 256, 0, stream>>>(kp1, kp2, ws);
  norm_apply<<<16, 256, 0, stream>>>(kp1, kp2, ws);

  for (int i = 0; i < RS_ITERS; i++) {
    gen_models<<<4, 256, 0, stream>>>(kp1, kp2, ws, i);
    verify_batch<<<NBATCH / 16, 256, 0, stream>>>(kp1, kp2, ws);
    select_best<<<1, 1024, 0, stream>>>(ws);
    compute_mask<<<16, 256, 0, stream>>>(kp1, kp2, ws);
    for (int lo = 0; lo < RS_LO; lo++) {
      for (int d = 0; d < RS_DLT; d++) {
        atwa_partial<<<32, 128, 0, stream>>>(ws);
        atwa_eig<<<1, 64, 0, stream>>>(ws);
        if (d < RS_DLT - 1) reweight<<<16, 256, 0, stream>>>(kp1, kp2, ws);
      }
      set_scalar<<<1, 1, 0, stream>>>(ws + S_TRSC, 0.f);
      verify_single<<<16, 256, 0, stream>>>(kp1, kp2, ws);
      polish_update<<<1, 1, 0, stream>>>(ws);
      post_lo<<<16, 256, 0, stream>>>(ws);
    }
    iter_update<<<1, 1, 0, stream>>>(ws, i);
    copy_best_inl<<<16, 256, 0, stream>>>(ws);
  }
  int nt = out_size > 0 ? out_size : (9 + NPTS);
  write_out<<<(nt + 255) / 256, 256, 0, stream>>>(ws, out, nt);
}